// SGC_20383914787293
// MI455X (gfx1250) — compile-verified
//
#include <hip/hip_runtime.h>

#define DF 64

typedef __attribute__((ext_vector_type(2))) float v2f;
typedef __attribute__((ext_vector_type(4))) float v4f;
typedef __attribute__((ext_vector_type(8))) float v8f;

// deg[i] = 1.0 (self loop)
__global__ void sgc_deg_init(float* __restrict__ deg, int n) {
  int i = blockIdx.x * blockDim.x + threadIdx.x;
  if (i < n) deg[i] = 1.0f;
}

// deg[dst[e]] += 1 for every edge
__global__ void sgc_deg_scatter(const int* __restrict__ dstI, float* __restrict__ deg, int e) {
  int i = blockIdx.x * blockDim.x + threadIdx.x;
  if (i < e) atomicAdd(&deg[dstI[i]], 1.0f);
}

// deg -> D^{-1/2} in place
__global__ void sgc_dinv(float* __restrict__ deg, int n) {
  int i = blockIdx.x * blockDim.x + threadIdx.x;
  if (i < n) {
    float d = deg[i];
    deg[i] = (d > 0.0f) ? rsqrtf(d) : 0.0f;
  }
}

// hout[i,:] = hin[i,:] * dinv[i]^2   (self-loop term; also fully initializes hout)
__global__ void sgc_hop_init(const float* __restrict__ hin, const float* __restrict__ dinv,
                             float* __restrict__ hout, int n) {
  int t = blockIdx.x * blockDim.x + threadIdx.x;   // n*16 threads, 4 floats each
  int i = t >> 4;
  int c = (t & 15) << 2;
  if (i >= n) return;
  float s = dinv[i];
  s *= s;
  v4f v = *(const v4f*)(hin + (size_t)i * DF + c);
  v4f o;
  o.x = v.x * s; o.y = v.y * s; o.z = v.z * s; o.w = v.w * s;
  *(v4f*)(hout + (size_t)i * DF + c) = o;
}

// hout[dst,:] += hin[src,:] * dinv[src]*dinv[dst] for every edge
__global__ void sgc_hop_scatter(const int* __restrict__ srcI, const int* __restrict__ dstI,
                                const float* __restrict__ dinv,
                                const float* __restrict__ hin, float* hout, int e) {
  int t = blockIdx.x * blockDim.x + threadIdx.x;   // e*16 threads, 4 floats each
  int ei = t >> 4;
  int c = (t & 15) << 2;
  if (ei >= e) return;
  int s = srcI[ei];
  int d = dstI[ei];
  float nrm = dinv[s] * dinv[d];
  v4f v = *(const v4f*)(hin + (size_t)s * DF + c);
  float* out = hout + (size_t)d * DF + c;
  atomicAdd(out + 0, v.x * nrm);
  atomicAdd(out + 1, v.y * nrm);
  atomicAdd(out + 2, v.z * nrm);
  atomicAdd(out + 3, v.w * nrm);
}

// y = h @ W^T + b, computed IN PLACE on hy ([n,64]); W is [64,64] row-major.
// One wave per 16-node tile; 4 16x16 accumulator tiles cover all 64 outputs.
// VGPR layouts per CDNA5 ISA 7.12.2:
//   A (16x4 f32, 2 VGPRs):  lane L holds A[M = L%16][K = kk + 2*(L/16) + {0,1}]
//   B (4x16 f32, 2 VGPRs):  lane L holds B[K = kk + 2*(L/16) + {0,1}][N = L%16]
//                           (B[k][nf] = W[nf][k], so same gather pattern on W rows)
//   C/D (16x16 f32, 8 VGPRs): lane L, vgpr r holds D[M = r + 8*(L/16)][N = L%16]
__global__ void __launch_bounds__(256) sgc_linear_wmma(
    float* hy, const float* __restrict__ W, const float* __restrict__ bias, int n) {
  const int lane = threadIdx.x & 31;
  const int wave = (blockIdx.x << 3) + (threadIdx.x >> 5);
  const int m_base = wave << 4;
  if (m_base >= n) return;            // wave-uniform: EXEC stays all-1s for WMMA

  const int nn = lane & 15;           // A row within tile / B-C-D column
  const int khalf = (lane >> 4) << 1; // +0 for lanes 0-15, +2 for lanes 16-31

  v8f acc0 = {0, 0, 0, 0, 0, 0, 0, 0};
  v8f acc1 = acc0, acc2 = acc0, acc3 = acc0;

  const float* hrow = hy + (size_t)(m_base + nn) * DF;
  const float* w0 = W + (size_t)(0 + nn) * DF;
  const float* w1 = W + (size_t)(16 + nn) * DF;
  const float* w2 = W + (size_t)(32 + nn) * DF;
  const float* w3 = W + (size_t)(48 + nn) * DF;

#pragma unroll
  for (int kk = 0; kk < DF; kk += 4) {
    v2f a;  a.x  = hrow[kk + khalf]; a.y  = hrow[kk + khalf + 1];
    v2f b0; b0.x = w0[kk + khalf];   b0.y = w0[kk + khalf + 1];
    v2f b1; b1.x = w1[kk + khalf];   b1.y = w1[kk + khalf + 1];
    v2f b2; b2.x = w2[kk + khalf];   b2.y = w2[kk + khalf + 1];
    v2f b3; b3.x = w3[kk + khalf];   b3.y = w3[kk + khalf + 1];
    acc0 = __builtin_amdgcn_wmma_f32_16x16x4_f32(false, a, false, b0, (short)0, acc0, false, false);
    acc1 = __builtin_amdgcn_wmma_f32_16x16x4_f32(false, a, false, b1, (short)0, acc1, false, false);
    acc2 = __builtin_amdgcn_wmma_f32_16x16x4_f32(false, a, false, b2, (short)0, acc2, false, false);
    acc3 = __builtin_amdgcn_wmma_f32_16x16x4_f32(false, a, false, b3, (short)0, acc3, false, false);
  }

  const int mo = (lane >> 4) << 3;    // D row offset: 0 or 8
  float bv0 = bias[0 + nn], bv1 = bias[16 + nn], bv2 = bias[32 + nn], bv3 = bias[48 + nn];
  float* yrow = hy + (size_t)(m_base + mo) * DF;
#pragma unroll
  for (int r = 0; r < 8; ++r) {
    float* yr = yrow + (size_t)r * DF;
    yr[0 + nn]  = acc0[r] + bv0;
    yr[16 + nn] = acc1[r] + bv1;
    yr[32 + nn] = acc2[r] + bv2;
    yr[48 + nn] = acc3[r] + bv3;
  }
}

extern "C" void kernel_launch(void* const* d_in, const int* in_sizes, int n_in,
                              void* d_out, int out_size, void* d_ws, size_t ws_size,
                              hipStream_t stream) {
  (void)n_in; (void)out_size; (void)ws_size;
  const float* x  = (const float*)d_in[0];   // [N,64]
  const int*   ei = (const int*)d_in[1];     // [2,E] row-major: [0]=src, [1]=dst
  const float* W  = (const float*)d_in[2];   // [64,64]
  const float* b  = (const float*)d_in[3];   // [64]
  const int N = in_sizes[0] / DF;
  const int E = in_sizes[1] / 2;
  const int* srcI = ei;
  const int* dstI = ei + E;

  // Workspace layout: dinv [N] floats, then h1 [N*64] floats (~26 MB total).
  float* dinv = (float*)d_ws;
  float* h1   = dinv + N;        // N*4 bytes offset keeps 16B alignment (N=100000)
  float* h2   = (float*)d_out;   // hop-2 result lives in d_out; GEMM runs in place

  const int TB = 256;
  sgc_deg_init<<<(N + TB - 1) / TB, TB, 0, stream>>>(dinv, N);
  sgc_deg_scatter<<<(E + TB - 1) / TB, TB, 0, stream>>>(dstI, dinv, E);
  sgc_dinv<<<(N + TB - 1) / TB, TB, 0, stream>>>(dinv, N);

  const long long tN = (long long)N * 16, tE = (long long)E * 16;
  // hop 1: x -> h1
  sgc_hop_init<<<(int)((tN + TB - 1) / TB), TB, 0, stream>>>(x, dinv, h1, N);
  sgc_hop_scatter<<<(int)((tE + TB - 1) / TB), TB, 0, stream>>>(srcI, dstI, dinv, x, h1, E);
  // hop 2: h1 -> h2 (d_out)
  sgc_hop_init<<<(int)((tN + TB - 1) / TB), TB, 0, stream>>>(h1, dinv, h2, N);
  sgc_hop_scatter<<<(int)((tE + TB - 1) / TB), TB, 0, stream>>>(srcI, dstI, dinv, h1, h2, E);

  // final linear via fp32 WMMA, in place on d_out
  const int waves  = (N + 15) / 16;
  const int blocks = (waves + 7) / 8;  // 8 waves (256 threads) per block
  sgc_linear_wmma<<<blocks, 256, 0, stream>>>(h2, W, b, N);
}